// ProbAttention_8340826488954
// MI455X (gfx1250) — compile-verified
//
#include <hip/hip_runtime.h>
#include <hip/hip_bf16.h>
#include <math.h>

// Problem constants fixed by the reference
#define BB 4
#define LL 2048
#define HH 8
#define DD 64
#define UPART 80
#define TOPU 40

typedef __attribute__((ext_vector_type(2))) float v2f;
typedef __attribute__((ext_vector_type(4))) float f32x4;
typedef __attribute__((ext_vector_type(8))) float v8f;

// ---------------------------------------------------------------------------
// Kernel 0: non-temporal zero fill of the 512MB output (store-bandwidth bound)
// ---------------------------------------------------------------------------
__global__ void pa_zero_out(f32x4* __restrict__ p, long n4) {
    long i = (long)blockIdx.x * blockDim.x + threadIdx.x;
    long stride = (long)gridDim.x * blockDim.x;
    f32x4 z = {0.f, 0.f, 0.f, 0.f};
    for (; i < n4; i += stride) {
        __builtin_nontemporal_store(z, p + i);
    }
}

// ---------------------------------------------------------------------------
// Kernel 1: M[b,h,l] = max_s(q.k_sample) - sum_s(q.k_sample)/L
// queries/keys layout: (B, L, H, D);  one thread per (b,h,l)
// ---------------------------------------------------------------------------
__global__ void pa_compute_M(const float* __restrict__ q,
                             const float* __restrict__ k,
                             const int* __restrict__ idxs,
                             float* __restrict__ Mout) {
    int t = blockIdx.x * blockDim.x + threadIdx.x;   // 0 .. B*H*L-1
    int l  = t % LL;
    int bh = t / LL;
    int h  = bh % HH;
    int b  = bh / HH;

    const f32x4* qp = (const f32x4*)(q + (((long)b * LL + l) * HH + h) * DD);
    f32x4 qr[16];
#pragma unroll
    for (int i = 0; i < 16; ++i) qr[i] = qp[i];

    float mx = -INFINITY;
    float sm = 0.f;
    for (int s = 0; s < UPART; ++s) {
        int idx = idxs[l * UPART + s];
        const f32x4* kp = (const f32x4*)(k + (((long)b * LL + idx) * HH + h) * DD);
        float acc = 0.f;
#pragma unroll
        for (int i = 0; i < 16; ++i) {
            f32x4 kv = kp[i];
            acc += qr[i].x * kv.x + qr[i].y * kv.y + qr[i].z * kv.z + qr[i].w * kv.w;
        }
        mx = fmaxf(mx, acc);
        sm += acc;
    }
    Mout[t] = mx - sm * (1.0f / (float)LL);
}

// ---------------------------------------------------------------------------
// Kernel 2: per (b,h) iterative argmax top-40 of 2048 (tie -> smaller index,
// matching jax top_k). One 256-thread block per (b,h).
// ---------------------------------------------------------------------------
__global__ void pa_topk(const float* __restrict__ Min, int* __restrict__ mtop) {
    __shared__ float vals[LL];
    __shared__ float rv[256];
    __shared__ int   ri[256];
    int bh = blockIdx.x;
    int t  = threadIdx.x;
#pragma unroll
    for (int i = 0; i < LL / 256; ++i)
        vals[i * 256 + t] = Min[(long)bh * LL + i * 256 + t];
    __syncthreads();

    for (int it = 0; it < TOPU; ++it) {
        float bv = -INFINITY;
        int   bi = LL;
#pragma unroll
        for (int i = 0; i < LL / 256; ++i) {
            int j = i * 256 + t;
            float v = vals[j];
            if (v > bv) { bv = v; bi = j; }
        }
        rv[t] = bv; ri[t] = bi;
        __syncthreads();
        for (int s = 128; s > 0; s >>= 1) {
            if (t < s) {
                if (rv[t + s] > rv[t] || (rv[t + s] == rv[t] && ri[t + s] < ri[t])) {
                    rv[t] = rv[t + s]; ri[t] = ri[t + s];
                }
            }
            __syncthreads();
        }
        if (t == 0) {
            mtop[bh * TOPU + it] = ri[0];
            vals[ri[0]] = -INFINITY;
        }
        __syncthreads();
    }
}

// ---------------------------------------------------------------------------
// Kernel 3: scores = (Q_reduce @ K^T) * 1/sqrt(D), written directly into the
// scattered output rows. Uses V_WMMA_F32_16X16X4_F32 (exact f32 path).
// Block = 128 threads = 4 waves; each wave owns one 16x16 output tile.
// Grid = BH * 3 rowtiles * 32 ngroups.
// ---------------------------------------------------------------------------
__global__ void pa_scores_wmma(const float* __restrict__ queries,
                               const float* __restrict__ keys,
                               const int* __restrict__ mtop,
                               float* __restrict__ out) {
    __shared__ float qtile[16 * DD];  // zero-padded Q_reduce row tile
    __shared__ int   rowl[16];        // scattered output row per tile row (-1 = pad)

    int ng = blockIdx.x & 31;          // 32 n-groups of 4 tiles
    int rt = (blockIdx.x >> 5) % 3;    // 3 row tiles (rows 0..47, 40 valid)
    int bh = blockIdx.x / 96;
    int h  = bh % HH;
    int b  = bh / HH;
    int t  = threadIdx.x;

    if (t < 16) {
        int mg = rt * 16 + t;
        rowl[t] = (mg < TOPU) ? mtop[bh * TOPU + mg] : -1;
    }
    __syncthreads();

    // Stage Q tile (16 x 64 f32, zero pad) into LDS: 256 float4, 2 per thread.
#pragma unroll
    for (int i = 0; i < 2; ++i) {
        int flat4 = t * 2 + i;          // 0..255
        int r  = flat4 >> 4;            // row 0..15
        int c4 = flat4 & 15;            // float4 col
        int rl = rowl[r];
        f32x4 v = {0.f, 0.f, 0.f, 0.f};
        if (rl >= 0)
            v = *(const f32x4*)(queries + (((long)b * LL + rl) * HH + h) * DD + c4 * 4);
        *(f32x4*)&qtile[r * DD + c4 * 4] = v;
    }
    __syncthreads();

    int wv   = t >> 5;
    int lane = t & 31;
    int half = lane >> 4;     // K-pair selector per ISA f32 16x4 layout
    int ln   = lane & 15;     // M for A-frag, N for B-frag
    int nb   = (ng * 4 + wv) * 16;

    const v2f* kp2 = (const v2f*)(keys + (((long)b * LL + (nb + ln)) * HH + h) * DD);
    const v2f* qp2 = (const v2f*)&qtile[ln * DD];

    v8f c = {};
#pragma unroll
    for (int kk = 0; kk < 16; ++kk) {   // K = 64 = 16 x 4
        v2f a = qp2[2 * kk + half];     // A[m=ln][4kk + 2*half + {0,1}]
        v2f bfrag = kp2[2 * kk + half]; // B[4kk + 2*half + {0,1}][n=nb+ln]
        c = __builtin_amdgcn_wmma_f32_16x16x4_f32(
                false, a, false, bfrag, (short)0, c, false, false);
    }

    // C/D layout: VGPR r -> M = r + 8*half, N = nb + ln. Scale = 1/sqrt(64).
#pragma unroll
    for (int r = 0; r < 8; ++r) {
        int ml = r + 8 * half;
        int rl = rowl[ml];
        if (rl >= 0) {
            long off = ((long)bh * LL + rl) * LL + (nb + ln);
            out[off] = c[r] * 0.125f;
        }
    }
}

// ---------------------------------------------------------------------------
// Kernel 4: in-place softmax over each scattered output row (2048 f32).
// One 256-thread block per (b,h,u); 8 elements per thread kept in registers.
// ---------------------------------------------------------------------------
__global__ void pa_softmax_rows(const int* __restrict__ mtop, float* __restrict__ out) {
    __shared__ float red[256];
    int bh = blockIdx.x / TOPU;
    int u  = blockIdx.x % TOPU;
    int rl = mtop[bh * TOPU + u];
    float* row = out + ((long)bh * LL + rl) * LL;
    int t = threadIdx.x;

    f32x4 v0 = *(f32x4*)&row[t * 4];
    f32x4 v1 = *(f32x4*)&row[1024 + t * 4];

    float mx = fmaxf(fmaxf(fmaxf(v0.x, v0.y), fmaxf(v0.z, v0.w)),
                     fmaxf(fmaxf(v1.x, v1.y), fmaxf(v1.z, v1.w)));
    red[t] = mx;
    __syncthreads();
    for (int s = 128; s > 0; s >>= 1) {
        if (t < s) red[t] = fmaxf(red[t], red[t + s]);
        __syncthreads();
    }
    float m = red[0];
    __syncthreads();

    f32x4 e0, e1;
    e0.x = expf(v0.x - m); e0.y = expf(v0.y - m);
    e0.z = expf(v0.z - m); e0.w = expf(v0.w - m);
    e1.x = expf(v1.x - m); e1.y = expf(v1.y - m);
    e1.z = expf(v1.z - m); e1.w = expf(v1.w - m);

    red[t] = (e0.x + e0.y + e0.z + e0.w) + (e1.x + e1.y + e1.z + e1.w);
    __syncthreads();
    for (int s = 128; s > 0; s >>= 1) {
        if (t < s) red[t] += red[t + s];
        __syncthreads();
    }
    float inv = 1.0f / red[0];

    e0.x *= inv; e0.y *= inv; e0.z *= inv; e0.w *= inv;
    e1.x *= inv; e1.y *= inv; e1.z *= inv; e1.w *= inv;
    *(f32x4*)&row[t * 4] = e0;
    *(f32x4*)&row[1024 + t * 4] = e1;
}

// ---------------------------------------------------------------------------
extern "C" void kernel_launch(void* const* d_in, const int* in_sizes, int n_in,
                              void* d_out, int out_size, void* d_ws, size_t ws_size,
                              hipStream_t stream) {
    const float* queries = (const float*)d_in[0];
    const float* keys    = (const float*)d_in[1];
    const int*   idxs    = (const int*)d_in[2];
    float* out = (float*)d_out;

    float* Mbuf = (float*)d_ws;                                     // B*H*L f32
    int*   mtop = (int*)((char*)d_ws + (size_t)BB * HH * LL * 4);   // B*H*40 i32

    // 512MB zero fill (dominant cost: HBM store bound, ~22us floor @ 23.3TB/s)
    long n4 = (long)out_size / 4;
    pa_zero_out<<<8192, 256, 0, stream>>>((f32x4*)d_out, n4);

    // sparsity measure M
    pa_compute_M<<<(BB * HH * LL) / 256, 256, 0, stream>>>(queries, keys, idxs, Mbuf);

    // top-40 per (b,h)
    pa_topk<<<BB * HH, 256, 0, stream>>>(Mbuf, mtop);

    // WMMA f32 GEMM: scores into scattered output rows
    pa_scores_wmma<<<BB * HH * 3 * 32, 128, 0, stream>>>(queries, keys, mtop, out);

    // in-place row softmax
    pa_softmax_rows<<<BB * HH * TOPU, 256, 0, stream>>>(mtop, out);
}